// EnsembleGNN_80625126080957
// MI455X (gfx1250) — compile-verified
//
#include <hip/hip_runtime.h>
#include <hip/hip_bf16.h>
#include <math.h>

// ---------------- problem constants (from reference) ----------------
#define NODES 50000
#define EDGES 400000
#define ETOT  (EDGES + NODES)   // edges + self loops
#define NEG_SLOPE 0.2f
#define BN_EPS 1e-5f

typedef __attribute__((ext_vector_type(2))) float v2f;
typedef __attribute__((ext_vector_type(8))) float v8f;

// float atomic max via int punning (works for any sign, init = -inf)
static __device__ __forceinline__ void atomicMaxF(float* addr, float v) {
  if (v >= 0.0f) atomicMax((int*)addr, __float_as_int(v));
  else           atomicMin((unsigned int*)addr, (unsigned int)__float_as_int(v));
}

// ---------------- generic fills ----------------
__global__ void fill_f32(float* __restrict__ p, float v, int n) {
  int i = blockIdx.x * blockDim.x + threadIdx.x;
  if (i < n) p[i] = v;
}

// ---------------- degree / normalization ----------------
__global__ void deg_edges(const int* __restrict__ ei, float* __restrict__ deg) {
  int e = blockIdx.x * blockDim.x + threadIdx.x;
  if (e < EDGES) atomicAdd(&deg[ei[EDGES + e]], 1.0f);
}
__global__ void dinv_k(const float* __restrict__ deg, float* __restrict__ dinv) {
  int i = blockIdx.x * blockDim.x + threadIdx.x;
  if (i < NODES) { float d = deg[i]; dinv[i] = d > 0.0f ? rsqrtf(d) : 0.0f; }
}

// ---------------- LDS-tiled WMMA f32 GEMM:  C[M,Nc] = A[M,K] @ B[K,Nc] ----------------
// Block = 4 waves (128 threads). Block tile = 64 rows x NT cols (NT = 16*CT).
// K-chunks of 32 are staged into LDS with coalesced float4 loads.
// LDS layouts are chosen so each WMMA fragment is ONE aligned ds_load_b64:
//   As[r][k]         : row stride 36 dwords (16B-aligned rows; 36r mod 64 distinct
//                      per lane -> conflict-free b64 fragment reads)
//   Bs[kp][2c+{0,1}] : K-pair interleaved {B[2kp][c], B[2kp+1][c]}, row stride
//                      NT*2+32 dwords (== 32 mod 64 -> the two lane-halves, which
//                      read rows kp and kp+1, hit disjoint bank halves)
// Fragment layouts per CDNA5 ISA 7.12.2:
//   A 16x4 : lanes 0-15 -> (M=l, K=k..k+1); lanes 16-31 -> K=k+2..k+3
//   B 4x16 : v[0]=row k / k+2 per lane half; v[1]=row k+1 / k+3
//   C/D    : v[r] holds rows r (lanes 0-15) and r+8 (lanes 16-31)
template <int CT>
__global__ __launch_bounds__(128)
void gemm_wmma_tiled(const float* __restrict__ A, const float* __restrict__ B,
                     float* __restrict__ C, int M, int K, int Nc) {
  constexpr int NT  = 16 * CT;       // block/wave column tile
  constexpr int KC  = 32;            // K chunk staged in LDS
  constexpr int ASD = KC + 4;        // 36: 16B-aligned, conflict-free
  constexpr int BSD = NT * 2 + 32;   // 160 / 96: == 32 mod 64
  __shared__ float As[64 * ASD];
  __shared__ float Bs[(KC / 2) * BSD];

  const int tid  = threadIdx.x;
  const int wave = tid >> 5;
  const int lane = tid & 31;
  const int half = lane >> 4;
  const int l16  = lane & 15;

  const int rowB = blockIdx.x * 64;          // block row base
  const int row0 = rowB + wave * 16;         // this wave's 16-row strip
  const int col0 = blockIdx.y * NT;
  const bool active = row0 < M;              // wave-uniform (M % 16 == 0)

  v8f zero = {};
  v8f acc[CT];
#pragma unroll
  for (int t = 0; t < CT; ++t) acc[t] = zero;

  for (int k0 = 0; k0 < K; k0 += KC) {
    // --- stage A chunk: rows rowB..rowB+63, cols k0..k0+31 (coalesced float4) ---
#pragma unroll
    for (int idx = tid; idx < 64 * (KC / 4); idx += 128) {
      int r  = idx >> 3;             // 8 float4 per row
      int c4 = (idx & 7) << 2;
      int gr = rowB + r;
      float4 v = (gr < M) ? *(const float4*)(A + (size_t)gr * K + k0 + c4)
                          : make_float4(0.f, 0.f, 0.f, 0.f);
      *(float4*)&As[r * ASD + c4] = v;   // 16B aligned (ASD % 4 == 0)
    }
    // --- stage B chunk, K-pair interleaved: Bs[kp][2c+{0,1}] = B[k0+2kp(+1)][col0+c] ---
#pragma unroll
    for (int idx = tid; idx < (KC / 2) * (NT / 4); idx += 128) {
      int kp = idx / (NT / 4);
      int cg = idx % (NT / 4);       // group of 4 columns
      const float* b0 = B + (size_t)(k0 + 2 * kp) * Nc + col0 + 4 * cg;
      float4 r0 = *(const float4*)b0;
      float4 r1 = *(const float4*)(b0 + Nc);
      float* dst = &Bs[kp * BSD + 8 * cg];   // 32B aligned
      *(float4*)(dst + 0) = make_float4(r0.x, r1.x, r0.y, r1.y);
      *(float4*)(dst + 4) = make_float4(r0.z, r1.z, r0.w, r1.w);
    }
    __syncthreads();

    if (active) {  // wave-uniform: EXEC all ones inside (WMMA requirement)
      const float* arow = &As[(wave * 16 + l16) * ASD];
#pragma unroll
      for (int kk = 0; kk < KC; kk += 4) {
        const int ka = kk + 2 * half;          // this half's K base
        v2f a = *(const v2f*)(arow + ka);      // one ds_load_b64
        const int kp = ka >> 1;                // interleaved B row
#pragma unroll
        for (int t = 0; t < CT; ++t) {
          v2f b = *(const v2f*)&Bs[kp * BSD + 2 * (t * 16 + l16)];  // one ds_load_b64
          acc[t] = __builtin_amdgcn_wmma_f32_16x16x4_f32(
              false, a, false, b, (short)0, acc[t], false, false);
        }
      }
    }
    __syncthreads();
  }

  if (active) {
#pragma unroll
    for (int t = 0; t < CT; ++t) {
#pragma unroll
      for (int r = 0; r < 8; ++r) {
        int m = row0 + r + 8 * half;
        C[(size_t)m * Nc + col0 + t * 16 + l16] = acc[t][r];
      }
    }
  }
}

// ---------------- GCN message scatter:  agg[d] += (hW[s]) * dinv[s]*dinv[d] ----------------
__global__ void gcn_scatter(const float* __restrict__ hW, const int* __restrict__ ei,
                            const float* __restrict__ dinv, float* __restrict__ agg, int fo) {
  const int c4 = fo >> 2;
  long tid = (long)blockIdx.x * blockDim.x + threadIdx.x;
  if (tid >= (long)ETOT * c4) return;
  int e  = (int)(tid / c4);
  int cc = (int)(tid % c4) << 2;
  int s, d;
  if (e < EDGES) { s = ei[e]; d = ei[EDGES + e]; } else { s = e - EDGES; d = s; }
  float nrm = dinv[s] * dinv[d];
  float4 v = *(const float4*)(hW + (size_t)s * fo + cc);
  float* p = agg + (size_t)d * fo + cc;
  atomicAdd(p + 0, v.x * nrm);
  atomicAdd(p + 1, v.y * nrm);
  atomicAdd(p + 2, v.z * nrm);
  atomicAdd(p + 3, v.w * nrm);
}

// ---------------- GCN node update: h = relu(bn(agg + b)) + (res [+ skip_b]) (in place) ----
__global__ void gcn_combine(float* __restrict__ agg, const float* __restrict__ b,
                            const float* __restrict__ g, const float* __restrict__ beta,
                            const float* __restrict__ res, const float* __restrict__ skip_b,
                            int fo) {
  long tid = (long)blockIdx.x * blockDim.x + threadIdx.x;
  if (tid >= (long)NODES * fo) return;
  int c = (int)(tid % fo);
  const float scale = rsqrtf(1.0f + BN_EPS);
  float h = (agg[tid] + b[c]) * (g[c] * scale) + beta[c];
  h = fmaxf(h, 0.0f);
  float r = res[tid] + (skip_b ? skip_b[c] : 0.0f);
  agg[tid] = h + r;
}

// ---------------- GAT attention coefficients a_s/a_d per (node, head) ----------------
__global__ void gat_coef(const float* __restrict__ hp, const float* __restrict__ a_src,
                         const float* __restrict__ a_dst, float* __restrict__ as_,
                         float* __restrict__ ad_, int H, int C) {
  int t = blockIdx.x * blockDim.x + threadIdx.x;
  if (t >= NODES * H) return;
  int n = t / H, h = t % H;
  const float* row = hp + ((size_t)n * H + h) * C;
  const float* sv = a_src + h * C;
  const float* dv = a_dst + h * C;
  float s1 = 0.f, s2 = 0.f;
  for (int c = 0; c < C; ++c) { float v = row[c]; s1 += v * sv[c]; s2 += v * dv[c]; }
  as_[t] = s1;
  ad_[t] = s2;
}

// ---------------- GAT edge pass 1: e = leaky_relu(a_s[s]+a_d[d]); segment max --------
__global__ void gat_edge1(const int* __restrict__ ei, const float* __restrict__ as_,
                          const float* __restrict__ ad_, float* __restrict__ eb,
                          float* __restrict__ emax, int H) {
  long t = (long)blockIdx.x * blockDim.x + threadIdx.x;
  if (t >= (long)ETOT * H) return;
  int h = (int)(t % H), e = (int)(t / H);
  int s, d;
  if (e < EDGES) { s = ei[e]; d = ei[EDGES + e]; } else { s = e - EDGES; d = s; }
  float ev = as_[(size_t)s * H + h] + ad_[(size_t)d * H + h];
  ev = ev > 0.0f ? ev : NEG_SLOPE * ev;
  eb[t] = ev;
  atomicMaxF(&emax[(size_t)d * H + h], ev);
}

// ---------------- GAT edge pass 2: ex = exp(e - emax[d]); segment sum ----------------
__global__ void gat_edge2(const int* __restrict__ ei, float* __restrict__ eb,
                          const float* __restrict__ emax, float* __restrict__ esum, int H) {
  long t = (long)blockIdx.x * blockDim.x + threadIdx.x;
  if (t >= (long)ETOT * H) return;
  int h = (int)(t % H), e = (int)(t / H);
  int d = (e < EDGES) ? ei[EDGES + e] : (e - EDGES);
  float ex = expf(eb[t] - emax[(size_t)d * H + h]);
  eb[t] = ex;
  atomicAdd(&esum[(size_t)d * H + h], ex);
}

// ---------------- GAT message scatter: o[d,h,:] += hp[s,h,:] * (ex / esum[d,h]) -------
__global__ void gat_scatter(const float* __restrict__ hp, const int* __restrict__ ei,
                            const float* __restrict__ eb, const float* __restrict__ esum,
                            float* __restrict__ o, int H, int C) {
  const int c4 = C >> 2;
  long tid = (long)blockIdx.x * blockDim.x + threadIdx.x;
  if (tid >= (long)ETOT * H * c4) return;
  int cc = (int)(tid % c4) << 2;
  long rem = tid / c4;
  int h = (int)(rem % H);
  int e = (int)(rem / H);
  int s, d;
  if (e < EDGES) { s = ei[e]; d = ei[EDGES + e]; } else { s = e - EDGES; d = s; }
  float alpha = eb[(size_t)e * H + h] / esum[(size_t)d * H + h];
  float4 v = *(const float4*)(hp + ((size_t)s * H + h) * C + cc);
  float* p = o + ((size_t)d * H + h) * C + cc;
  atomicAdd(p + 0, v.x * alpha);
  atomicAdd(p + 1, v.y * alpha);
  atomicAdd(p + 2, v.z * alpha);
  atomicAdd(p + 3, v.w * alpha);
}

// ---------------- GAT node update: h = relu(bn(o + b)) (in place) ----------------
__global__ void gat_combine(float* __restrict__ o, const float* __restrict__ b,
                            const float* __restrict__ g, const float* __restrict__ beta,
                            int fo) {
  long tid = (long)blockIdx.x * blockDim.x + threadIdx.x;
  if (tid >= (long)NODES * fo) return;
  int c = (int)(tid % fo);
  const float scale = rsqrtf(1.0f + BN_EPS);
  float h = (o[tid] + b[c]) * (g[c] * scale) + beta[c];
  o[tid] = fmaxf(h, 0.0f);
}

// ---------------- MLP head 32 -> 16 (relu) -> 2, log_softmax ----------------
__global__ void classifier_k(const float* __restrict__ h, const float* __restrict__ W1,
                             const float* __restrict__ b1, const float* __restrict__ W2,
                             const float* __restrict__ b2, float* __restrict__ out) {
  int n = blockIdx.x * blockDim.x + threadIdx.x;
  if (n >= NODES) return;
  const float* hr = h + (size_t)n * 32;
  float o0 = b2[0], o1 = b2[1];
#pragma unroll 4
  for (int j = 0; j < 16; ++j) {
    float s = b1[j];
    for (int i = 0; i < 32; ++i) s += hr[i] * W1[i * 16 + j];
    s = fmaxf(s, 0.0f);
    o0 += s * W2[j * 2 + 0];
    o1 += s * W2[j * 2 + 1];
  }
  float m = fmaxf(o0, o1);
  float lse = m + logf(expf(o0 - m) + expf(o1 - m));
  out[(size_t)n * 2 + 0] = o0 - lse;
  out[(size_t)n * 2 + 1] = o1 - lse;
}

// ---------------- ensemble: softmax(ens_w)-weighted sum of log-probs ----------------
__global__ void ensemble_k(const float* __restrict__ og, const float* __restrict__ oa,
                           const float* __restrict__ ew, float* __restrict__ out) {
  int i = blockIdx.x * blockDim.x + threadIdx.x;
  if (i >= NODES * 2) return;
  float e0 = ew[0], e1 = ew[1];
  float m = fmaxf(e0, e1);
  float x0 = expf(e0 - m), x1 = expf(e1 - m);
  float inv = 1.0f / (x0 + x1);
  out[i] = (x0 * inv) * og[i] + (x1 * inv) * oa[i];
}

// ---------------- host side ----------------
static inline unsigned blocks_for(long n, int tb) { return (unsigned)((n + tb - 1) / tb); }

static void launch_gemm(const float* A, const float* B, float* C, int M, int K, int Nc,
                        hipStream_t s) {
  unsigned gx = (unsigned)((M + 63) / 64);
  if (Nc % 64 == 0) {
    gemm_wmma_tiled<4><<<dim3(gx, Nc / 64), 128, 0, s>>>(A, B, C, M, K, Nc);
  } else {  // Nc % 32 == 0 (all dims here are multiples of 32)
    gemm_wmma_tiled<2><<<dim3(gx, Nc / 32), 128, 0, s>>>(A, B, C, M, K, Nc);
  }
}

extern "C" void kernel_launch(void* const* d_in, const int* in_sizes, int n_in,
                              void* d_out, int out_size, void* d_ws, size_t ws_size,
                              hipStream_t stream) {
  (void)in_sizes; (void)n_in; (void)out_size; (void)ws_size;
  const int TB = 256;

  // ---- inputs (setup_inputs() dict insertion order, recursively flattened) ----
  const float* x  = (const float*)d_in[0];   // [N,128]
  const int*   ei = (const int*)d_in[1];     // [2,E] (row0=src, row1=dst)
  int p = 2;
  // GCN layer 0 (128->128, no skip)
  const float *g0W = (const float*)d_in[p++], *g0b = (const float*)d_in[p++],
              *g0g = (const float*)d_in[p++], *g0be = (const float*)d_in[p++];
  // GCN layer 1 (128->64, skip)
  const float *g1W = (const float*)d_in[p++], *g1b = (const float*)d_in[p++],
              *g1g = (const float*)d_in[p++], *g1be = (const float*)d_in[p++],
              *g1sW = (const float*)d_in[p++], *g1sb = (const float*)d_in[p++];
  // GCN layer 2 (64->32, skip)
  const float *g2W = (const float*)d_in[p++], *g2b = (const float*)d_in[p++],
              *g2g = (const float*)d_in[p++], *g2be = (const float*)d_in[p++],
              *g2sW = (const float*)d_in[p++], *g2sb = (const float*)d_in[p++];
  // GAT layer 0 (128 -> 8x128)
  const float *a0W = (const float*)d_in[p++], *a0as = (const float*)d_in[p++],
              *a0ad = (const float*)d_in[p++], *a0b = (const float*)d_in[p++],
              *a0g = (const float*)d_in[p++], *a0be = (const float*)d_in[p++];
  // GAT layer 1 (1024 -> 4x64)
  const float *a1W = (const float*)d_in[p++], *a1as = (const float*)d_in[p++],
              *a1ad = (const float*)d_in[p++], *a1b = (const float*)d_in[p++],
              *a1g = (const float*)d_in[p++], *a1be = (const float*)d_in[p++];
  // GAT layer 2 (256 -> 1x32)
  const float *a2W = (const float*)d_in[p++], *a2as = (const float*)d_in[p++],
              *a2ad = (const float*)d_in[p++], *a2b = (const float*)d_in[p++],
              *a2g = (const float*)d_in[p++], *a2be = (const float*)d_in[p++];
  // classifiers + ensemble weights
  const float *cgW1 = (const float*)d_in[p++], *cgb1 = (const float*)d_in[p++],
              *cgW2 = (const float*)d_in[p++], *cgb2 = (const float*)d_in[p++];
  const float *caW1 = (const float*)d_in[p++], *cab1 = (const float*)d_in[p++],
              *caW2 = (const float*)d_in[p++], *cab2 = (const float*)d_in[p++];
  const float *ensw = (const float*)d_in[p++];

  // ---- workspace layout ----
  float* ws = (float*)d_ws;
  size_t off = 0;
  float* T   = ws + off; off += (size_t)NODES * 1024;  // gemm output / hp
  float* O1  = ws + off; off += (size_t)NODES * 1024;  // agg ping
  float* O2  = ws + off; off += (size_t)NODES * 1024;  // agg pong
  float* R   = ws + off; off += (size_t)NODES * 128;   // gcn residual
  float* EB  = ws + off; off += (size_t)ETOT * 8;      // per-edge e / exp(e)
  float* DEG = ws + off; off += NODES;
  float* DNV = ws + off; off += NODES;
  float* AS  = ws + off; off += (size_t)NODES * 8;
  float* AD  = ws + off; off += (size_t)NODES * 8;
  float* EMX = ws + off; off += (size_t)NODES * 8;
  float* ESM = ws + off; off += (size_t)NODES * 8;
  float* OG  = ws + off; off += (size_t)NODES * 2;     // log-probs GCN branch
  float* OA  = ws + off; off += (size_t)NODES * 2;     // log-probs GAT branch

  // ---- degrees / symmetric norm (self loop included via init 1.0) ----
  fill_f32<<<blocks_for(NODES, TB), TB, 0, stream>>>(DEG, 1.0f, NODES);
  deg_edges<<<blocks_for(EDGES, TB), TB, 0, stream>>>(ei, DEG);
  dinv_k<<<blocks_for(NODES, TB), TB, 0, stream>>>(DEG, DNV);

  // =================== GCN branch ===================
  // L0: 128->128, res = x
  launch_gemm(x, g0W, T, NODES, 128, 128, stream);
  hipMemsetAsync(O1, 0, (size_t)NODES * 128 * 4, stream);
  gcn_scatter<<<blocks_for((long)ETOT * 32, TB), TB, 0, stream>>>(T, ei, DNV, O1, 128);
  gcn_combine<<<blocks_for((long)NODES * 128, TB), TB, 0, stream>>>(O1, g0b, g0g, g0be, x, nullptr, 128);
  // L1: 128->64, res = h @ skip_W + skip_b
  launch_gemm(O1, g1sW, R, NODES, 128, 64, stream);
  launch_gemm(O1, g1W,  T, NODES, 128, 64, stream);
  hipMemsetAsync(O2, 0, (size_t)NODES * 64 * 4, stream);
  gcn_scatter<<<blocks_for((long)ETOT * 16, TB), TB, 0, stream>>>(T, ei, DNV, O2, 64);
  gcn_combine<<<blocks_for((long)NODES * 64, TB), TB, 0, stream>>>(O2, g1b, g1g, g1be, R, g1sb, 64);
  // L2: 64->32
  launch_gemm(O2, g2sW, R, NODES, 64, 32, stream);
  launch_gemm(O2, g2W,  T, NODES, 64, 32, stream);
  hipMemsetAsync(O1, 0, (size_t)NODES * 32 * 4, stream);
  gcn_scatter<<<blocks_for((long)ETOT * 8, TB), TB, 0, stream>>>(T, ei, DNV, O1, 32);
  gcn_combine<<<blocks_for((long)NODES * 32, TB), TB, 0, stream>>>(O1, g2b, g2g, g2be, R, g2sb, 32);
  classifier_k<<<blocks_for(NODES, TB), TB, 0, stream>>>(O1, cgW1, cgb1, cgW2, cgb2, OG);

  // =================== GAT branch ===================
  struct Cfg { int H, C; };
  const Cfg cfg[3] = {{8, 128}, {4, 64}, {1, 32}};
  const float* gatW[3]  = {a0W, a1W, a2W};
  const float* gatAS[3] = {a0as, a1as, a2as};
  const float* gatAD[3] = {a0ad, a1ad, a2ad};
  const float* gatB[3]  = {a0b, a1b, a2b};
  const float* gatG[3]  = {a0g, a1g, a2g};
  const float* gatBe[3] = {a0be, a1be, a2be};
  const float* hin = x;
  int fi = 128;
  float* obuf[3] = {O2, O1, O2};  // ping-pong aggregation/output buffers
  for (int L = 0; L < 3; ++L) {
    const int H = cfg[L].H, C = cfg[L].C, fo = H * C;
    float* O = obuf[L];
    launch_gemm(hin, gatW[L], T, NODES, fi, fo, stream);                       // hp = h @ W
    gat_coef<<<blocks_for((long)NODES * H, TB), TB, 0, stream>>>(T, gatAS[L], gatAD[L], AS, AD, H, C);
    fill_f32<<<blocks_for((long)NODES * H, TB), TB, 0, stream>>>(EMX, -INFINITY, NODES * H);
    hipMemsetAsync(ESM, 0, (size_t)NODES * H * 4, stream);
    gat_edge1<<<blocks_for((long)ETOT * H, TB), TB, 0, stream>>>(ei, AS, AD, EB, EMX, H);
    gat_edge2<<<blocks_for((long)ETOT * H, TB), TB, 0, stream>>>(ei, EB, EMX, ESM, H);
    hipMemsetAsync(O, 0, (size_t)NODES * fo * 4, stream);
    gat_scatter<<<blocks_for((long)ETOT * H * (C >> 2), TB), TB, 0, stream>>>(T, ei, EB, ESM, O, H, C);
    gat_combine<<<blocks_for((long)NODES * fo, TB), TB, 0, stream>>>(O, gatB[L], gatG[L], gatBe[L], fo);
    hin = O;
    fi = fo;   // last layer: H=1, mean over 1 head == concat
  }
  classifier_k<<<blocks_for(NODES, TB), TB, 0, stream>>>(obuf[2], caW1, cab1, caW2, cab2, OA);

  // =================== ensemble ===================
  ensemble_k<<<blocks_for((long)NODES * 2, TB), TB, 0, stream>>>(OG, OA, ensw, (float*)d_out);
}